// DiffJPEG_21311627723254
// MI455X (gfx1250) — compile-verified
//
#include <hip/hip_runtime.h>
#include <math.h>

typedef __attribute__((ext_vector_type(2))) float v2f;
typedef __attribute__((ext_vector_type(8))) float v8f;

__device__ __constant__ float c_YTAB[64] = {
  16,11,10,16,24,40,51,61,
  12,12,14,19,26,58,60,55,
  14,13,16,24,40,57,69,56,
  14,17,22,29,51,87,80,62,
  18,22,37,56,68,109,103,77,
  24,35,55,64,81,104,113,92,
  49,64,78,87,103,121,120,101,
  72,92,95,98,112,100,103,99};

__device__ __constant__ float c_CTAB[64] = {
  17,18,24,47,99,99,99,99,
  18,21,26,66,99,99,99,99,
  24,26,56,99,99,99,99,99,
  47,66,99,99,99,99,99,99,
  99,99,99,99,99,99,99,99,
  99,99,99,99,99,99,99,99,
  99,99,99,99,99,99,99,99,
  99,99,99,99,99,99,99,99};

#define INV_SQRT2 0.70710678118654752440f

__device__ __forceinline__ float cos8(int x, int u) {
  return cosf((float)((2 * x + 1) * u) * (3.14159265358979323846f / 16.0f));
}
// blockdiag(C, C) 16x16, C[x][u] = cos((2x+1)u*pi/16)
__device__ __forceinline__ float B16v(int i, int j) {
  return ((i >> 3) == (j >> 3)) ? cos8(i & 7, j & 7) : 0.0f;
}

__device__ __forceinline__ v8f wmma16x16(const v2f A[4], const v2f B[4]) {
  v8f acc = {0.f, 0.f, 0.f, 0.f, 0.f, 0.f, 0.f, 0.f};
#pragma unroll
  for (int kk = 0; kk < 4; ++kk) {
    acc = __builtin_amdgcn_wmma_f32_16x16x4_f32(false, A[kk], false, B[kk],
                                                (short)0, acc, false, false);
  }
  return acc;
}

__device__ __forceinline__ void lds_sync() {
  asm volatile("s_wait_dscnt 0" ::: "memory");
}

// Store a C/D-layout (v8f) 16x16 matrix into LDS (row-major, stride 17 to dodge bank conflicts)
__device__ __forceinline__ void stC(float* S, const v8f& v, int hi, int n) {
#pragma unroll
  for (int r = 0; r < 8; ++r) S[(r + 8 * hi) * 17 + n] = v[r];
}
// Load as WMMA A-operand slices: slice kk, lane: M=n, K = 4kk+2hi (+1)
__device__ __forceinline__ void ldA(const float* S, v2f A[4], int hi, int n) {
#pragma unroll
  for (int kk = 0; kk < 4; ++kk) {
    int k0 = 4 * kk + 2 * hi;
    A[kk] = (v2f){S[n * 17 + k0], S[n * 17 + k0 + 1]};
  }
}
// Load as WMMA B-operand slices: slice kk, lane: N=n, K = 4kk+2hi (+1)
__device__ __forceinline__ void ldB(const float* S, v2f B[4], int hi, int n) {
#pragma unroll
  for (int kk = 0; kk < 4; ++kk) {
    int k0 = 4 * kk + 2 * hi;
    B[kk] = (v2f){S[k0 * 17 + n], S[(k0 + 1) * 17 + n]};
  }
}

// DCT -> quant (diff-round) -> dequant -> IDCT of one 16x16 block-diag-packed matrix.
// inB: input (minus 128) in B-operand layout (reg rr -> row 4*(rr/2)+(rr%2)+2*hi, col n)
// P[kk] = A-op of blockdiag(C^T)  (also equals B-op of blockdiag(C))
// Q[kk] = A-op of blockdiag(C)    (also equals B-op of blockdiag(C^T))
// Returns decoded spatial matrix (+128) in C/D layout.
__device__ __forceinline__ v8f jpeg_pipe(float* S, const float inB[8],
                                         const v2f P[4], const v2f Q[4],
                                         const float* tab, float fq,
                                         int hi, int n) {
  // mm1: T1 = C^T * X
  v2f Bx[4];
#pragma unroll
  for (int kk = 0; kk < 4; ++kk) Bx[kk] = (v2f){inB[2 * kk], inB[2 * kk + 1]};
  v8f T1 = wmma16x16(P, Bx);

  lds_sync();
  stC(S, T1, hi, n);
  lds_sync();
  v2f Aop[4];
  ldA(S, Aop, hi, n);
  // mm2: D = T1 * C   (raw 2D DCT, no alpha scaling yet)
  v8f D = wmma16x16(Aop, P);

  // quantize with differentiable round, then dequantize and fold in alpha_o for IDCT
  const float av = ((n & 7) == 0) ? INV_SQRT2 : 1.0f;
  v8f M = {0.f, 0.f, 0.f, 0.f, 0.f, 0.f, 0.f, 0.f};
#pragma unroll
  for (int r = 0; r < 8; ++r) {
    const float au = (r == 0) ? INV_SQRT2 : 1.0f;  // u = (r+8hi)%8 == r
    const float sc = 0.25f * au * av;              // dct scale
    const float qt = tab[r * 8 + (n & 7)] * fq;    // quant step
    float t = (D[r] * sc) / qt;
    float rnd = rintf(t);
    float dr = t - rnd;
    float q = rnd + dr * dr * dr;                  // diff_round
    M[r] = q * qt * au * av;                       // dequant * alpha_o
  }

  lds_sync();
  stC(S, M, hi, n);
  lds_sync();
  v2f Bop[4];
  ldB(S, Bop, hi, n);
  // mm3: T3 = C * M
  v8f T3 = wmma16x16(Q, Bop);

  lds_sync();
  stC(S, T3, hi, n);
  lds_sync();
  ldA(S, Aop, hi, n);
  // mm4: O = T3 * C^T
  v8f O = wmma16x16(Aop, Q);
#pragma unroll
  for (int r = 0; r < 8; ++r) O[r] = 0.25f * O[r] + 128.0f;
  return O;
}

__global__ __launch_bounds__(256) void diffjpeg_kernel(
    const float* __restrict__ x, const float* __restrict__ quality,
    float* __restrict__ out) {
  __shared__ float sh[8 * 272];  // 16x17 f32 scratch per wave
  const int lane = threadIdx.x & 31;
  const int wv = threadIdx.x >> 5;
  float* S = sh + wv * 272;

  const int tile = blockIdx.x * 8 + wv;  // 32768 tiles total
  const int img = tile >> 10;            // 1024 tiles per image (32x32)
  const int tt = tile & 1023;
  const int ty = tt >> 5, tx = tt & 31;
  const int hi = lane >> 4;   // lane half (K/M-row offset selector)
  const int n = lane & 15;    // column within tile

  // per-image quality factor
  const float qv = quality[img];
  const float fq = (qv < 50.0f ? 5000.0f / qv : 200.0f - 2.0f * qv) * 0.01f;

  // constant WMMA operands (16 VGPRs total)
  v2f P[4], Q[4];
#pragma unroll
  for (int kk = 0; kk < 4; ++kk) {
    int k0 = 4 * kk + 2 * hi;
    P[kk] = (v2f){B16v(k0, n), B16v(k0 + 1, n)};  // blockdiag(C^T) as A-op
    Q[kk] = (v2f){B16v(n, k0), B16v(n, k0 + 1)};  // blockdiag(C)   as A-op
  }

  // ---- load RGB tile directly in B-operand layout ----
  const int gx = tx * 16 + n;
  const size_t imgBase = (size_t)img * 3u * 512u * 512u;
  float Rv[8], Gv[8], Bv[8];
#pragma unroll
  for (int rr = 0; rr < 8; ++rr) {
    int row = 4 * (rr >> 1) + (rr & 1) + 2 * hi;
    size_t off = imgBase + (size_t)(ty * 16 + row) * 512 + gx;
    Rv[rr] = x[off] * 255.0f;
    Gv[rr] = x[off + 512 * 512] * 255.0f;
    Bv[rr] = x[off + 2 * 512 * 512] * 255.0f;
  }

  // ---- RGB -> YCbCr (B-layout, elementwise) ----
  float ym[8], cbB[8], crB[8];
#pragma unroll
  for (int rr = 0; rr < 8; ++rr) {
    ym[rr] = 0.299f * Rv[rr] + 0.587f * Gv[rr] + 0.114f * Bv[rr] - 128.0f;
    cbB[rr] = -0.168736f * Rv[rr] - 0.331264f * Gv[rr] + 0.5f * Bv[rr] + 128.0f;
    crB[rr] = 0.5f * Rv[rr] - 0.418688f * Gv[rr] - 0.081312f * Bv[rr] + 128.0f;
  }

  // ---- 2x2 chroma downsample: vertical pairs in-register, horizontal via shfl_xor ----
  float ucb[4], ucr[4];
#pragma unroll
  for (int k = 0; k < 4; ++k) {
    float tb = cbB[2 * k] + cbB[2 * k + 1];
    float tr = crB[2 * k] + crB[2 * k + 1];
    ucb[k] = tb + __shfl_xor(tb, 1, 32);  // DS row i = 2k+hi, col pair at lane&~1
    ucr[k] = tr + __shfl_xor(tr, 1, 32);
  }

  // ---- pack Cb (rows 0-7, cols 0-7) and Cr (rows 8-15, cols 8-15) block-diag, B-layout ----
  float chm[8];
#pragma unroll
  for (int rr = 0; rr < 8; ++rr) {
    int kk = rr >> 1, j = rr & 1;  // target row = 4kk + j + 2hi
    int sl = j * 16 + 2 * (n & 7); // source lane holding DS(row, col)
    float lo, hp;
    if (kk < 2) {  // Cb rows 0..7 -> u reg index = 2kk + hi
      lo = __shfl(ucb[2 * kk], sl, 32);
      hp = __shfl(ucb[2 * kk + 1], sl, 32);
    } else {       // Cr rows 8..15
      lo = __shfl(ucr[2 * (kk - 2)], sl, 32);
      hp = __shfl(ucr[2 * (kk - 2) + 1], sl, 32);
    }
    float dsv = (hi ? hp : lo) * 0.25f - 128.0f;
    bool valid = (kk < 2) ? (n < 8) : (n >= 8);
    chm[rr] = valid ? dsv : 0.0f;  // off-diagonal quadrants are zero
  }

  // ---- DCT/quant/dequant/IDCT chains (8 x 16x16x16 f32 matmuls = 32 WMMA) ----
  v8f Yd = jpeg_pipe(S, ym, P, Q, c_YTAB, fq, hi, n);
  v8f Cd = jpeg_pipe(S, chm, P, Q, c_CTAB, fq, hi, n);

  // ---- upsample chroma, YCbCr -> RGB, clip, store (C/D layout: row = r+8hi, col = n) ----
#pragma unroll
  for (int r = 0; r < 8; ++r) {
    int row = r + 8 * hi;
    int c2 = n >> 1;
    // Cb spatial (row/2, col/2): C-layout reg row/2 (+4 for hi half), lane c2
    float cb_l = __shfl(Cd[r >> 1], c2, 32);
    float cb_h = __shfl(Cd[(r >> 1) + 4], c2, 32);
    // Cr spatial at matrix (8+row/2, 8+col/2): same regs, lane 24+c2
    float cr_l = __shfl(Cd[r >> 1], 24 + c2, 32);
    float cr_h = __shfl(Cd[(r >> 1) + 4], 24 + c2, 32);
    float cb = (hi ? cb_h : cb_l) - 128.0f;
    float cr = (hi ? cr_h : cr_l) - 128.0f;
    float Yv = Yd[r];
    float Ro = Yv + 1.402f * cr;
    float Go = Yv - 0.344136f * cb - 0.714136f * cr;
    float Bo = Yv + 1.772f * cb;
    Ro = fminf(fmaxf(Ro, 0.0f), 255.0f) * (1.0f / 255.0f);
    Go = fminf(fmaxf(Go, 0.0f), 255.0f) * (1.0f / 255.0f);
    Bo = fminf(fmaxf(Bo, 0.0f), 255.0f) * (1.0f / 255.0f);
    size_t off = imgBase + (size_t)(ty * 16 + row) * 512 + gx;
    out[off] = Ro;
    out[off + 512 * 512] = Go;
    out[off + 2 * 512 * 512] = Bo;
  }
}

extern "C" void kernel_launch(void* const* d_in, const int* in_sizes, int n_in,
                              void* d_out, int out_size, void* d_ws,
                              size_t ws_size, hipStream_t stream) {
  (void)in_sizes; (void)n_in; (void)d_ws; (void)ws_size; (void)out_size;
  const float* x = (const float*)d_in[0];
  const float* quality = (const float*)d_in[1];
  float* out = (float*)d_out;
  // 32 images * (32x32) 16x16-tiles = 32768 tiles; 8 waves (tiles) per 256-thread block
  dim3 grid(4096), block(256);
  hipLaunchKernelGGL(diffjpeg_kernel, grid, block, 0, stream, x, quality, out);
}